// TGNPLMemory_32615981645895
// MI455X (gfx1250) — compile-verified
//
#include <hip/hip_runtime.h>
#include <hip/hip_bf16.h>
#include <climits>
#include <math.h>

#define N_NODES 100000
#define N_EVENTS 100000
#define MEM_D 128
#define RAW_D 64
#define TIME_D 64
#define MSG_D 512          // 3*MEM_D + RAW_D + TIME_D
#define KTOT 640           // MSG_D + MEM_D (fused [aggr | h])
#define GATES 384          // 3*MEM_D
#define LDSTR 644          // 640 + 4 pad -> conflict-free column reads (stride mod 64 = 4)
#define ROWS 32            // node rows per block (2 row-tiles per wave -> B reuse x2)

typedef __attribute__((ext_vector_type(2))) float v2f;
typedef __attribute__((ext_vector_type(8))) float v8f;

// ---------------------------------------------------------------- init scratch
__global__ void k_init(float* __restrict__ sumMsg, int* __restrict__ cnt,
                       int* __restrict__ tmax, float* __restrict__ loss) {
    long long i = (long long)blockIdx.x * blockDim.x + threadIdx.x;
    long long stride = (long long)gridDim.x * blockDim.x;
    const long long total = (long long)N_NODES * MSG_D;
    for (long long j = i; j < total; j += stride) sumMsg[j] = 0.0f;
    for (long long j = i; j < N_NODES; j += stride) { cnt[j] = 0; tmax[j] = INT_MIN; }
    if (i == 0) loss[0] = 0.0f;
}

// ------------------------------------------------- weight transpose / packing
// Bp is K-major, pair-packed: row kk/2 holds, for every output o in [0,384),
// the float2 { Wcat[o][kk], Wcat[o][kk+1] } at element index o.
__global__ void k_wprep(const float* __restrict__ Wih, const float* __restrict__ Whh,
                        float* __restrict__ Bp) {
    int i = blockIdx.x * blockDim.x + threadIdx.x;
    if (i >= KTOT * GATES) return;
    int k = i / GATES;
    int o = i - k * GATES;
    float v = (k < MSG_D) ? Wih[o * MSG_D + k] : Whh[o * MEM_D + (k - MSG_D)];
    Bp[(k >> 1) * (2 * GATES) + o * 2 + (k & 1)] = v;
}

// --------------------------------------------------------- event scatter-add
// One 512-thread block per event; thread j computes msg component j and
// scatter-adds it to sum_msg[src], sum_msg[dst], sum_msg[prod].
__global__ void __launch_bounds__(512)
k_events(const int* __restrict__ src, const int* __restrict__ dst,
         const int* __restrict__ prod, const int* __restrict__ t,
         const float* __restrict__ raw, const float* __restrict__ mem,
         const int* __restrict__ lastu, const float* __restrict__ Wt,
         const float* __restrict__ bt,
         float* __restrict__ sumMsg, int* __restrict__ cnt, int* __restrict__ tmax) {
    int e = blockIdx.x;
    int j = threadIdx.x;
    int s = src[e], d = dst[e], p = prod[e], tv = t[e];

    float val;
    if (j < 128)        val = mem[s * MEM_D + j];
    else if (j < 256)   val = mem[d * MEM_D + (j - 128)];
    else if (j < 384)   val = mem[p * MEM_D + (j - 256)];
    else if (j < 448)   val = raw[e * RAW_D + (j - 384)];
    else {
        float trel = (float)(tv - lastu[s]);
        int q = j - 448;
        val = cosf(trel * Wt[q] + bt[q]);
    }
    unsafeAtomicAdd(&sumMsg[s * MSG_D + j], val);
    unsafeAtomicAdd(&sumMsg[d * MSG_D + j], val);
    unsafeAtomicAdd(&sumMsg[p * MSG_D + j], val);
    if (j == 0) {
        atomicAdd(&cnt[s], 1); atomicAdd(&cnt[d], 1); atomicAdd(&cnt[p], 1);
        atomicMax(&tmax[s], tv); atomicMax(&tmax[d], tv); atomicMax(&tmax[p], tv);
    }
}

__device__ __forceinline__ float sigmoidf(float x) {
    return 1.0f / (1.0f + __expf(-x));
}

#define WMMA_F32(A, B, C) \
    __builtin_amdgcn_wmma_f32_16x16x4_f32(false, (A), false, (B), (short)0, (C), false, false)

// ---------------------------------------------- fused GEMM (WMMA f32) + GRU
// Block = 256 threads (8 wave32s) handles 32 node rows (two 16-row tiles).
// Wave w owns memory dims [16w, 16w+16): accumulates r-, z-, i_n-, h_n-gate
// tiles with v_wmma_f32_16x16x4_f32 over K = 640 ([aggr(512) | h(128)]).
// Each B fragment is reused for both row-tiles (halves L2 weight traffic).
__global__ void __launch_bounds__(256)
k_gru(const float* __restrict__ sumMsg, const int* __restrict__ cnt,
      const int* __restrict__ tmax, const float* __restrict__ mem,
      const int* __restrict__ lastu, const float* __restrict__ initm,
      const float* __restrict__ Bp, const float* __restrict__ bih,
      const float* __restrict__ bhh,
      float* __restrict__ outMem, float* __restrict__ outLast,
      float* __restrict__ loss) {
    __shared__ __align__(16) float Ads[ROWS * LDSTR];   // 32*644*4 = 82,432 B (CDNA5 LDS)
    __shared__ float invS[ROWS];
    __shared__ float hasS[ROWS];

    int tid = threadIdx.x;
    int nodeBase = blockIdx.x * ROWS;

    // Warm the caches with this block's sum_msg tile (64 KB = 256 threads x 256 B)
    {
        const char* pfb = (const char*)(sumMsg + (size_t)nodeBase * MSG_D);
        __builtin_prefetch(pfb + tid * 256, 0, 0);
        __builtin_prefetch(pfb + tid * 256 + 128, 0, 0);
    }

    if (tid < ROWS) {
        int node = nodeBase + tid;
        int c = cnt[node];
        hasS[tid] = (c > 0) ? 1.0f : 0.0f;
        invS[tid] = 1.0f / fmaxf((float)c, 1.0f);
        outLast[node] = (c > 0) ? (float)tmax[node] : (float)lastu[node];
    }
    __syncthreads();

    // Load A-tile: columns [0,512) = aggr = sum_msg * 1/max(count,1),
    //              columns [512,640) = h = (last_update==-1 ? init : mem).
    for (int i = tid; i < ROWS * KTOT; i += 256) {
        int row = i / KTOT;
        int col = i - row * KTOT;
        int node = nodeBase + row;
        float v;
        if (col < MSG_D) {
            v = sumMsg[node * MSG_D + col] * invS[row];
        } else {
            int dc = col - MSG_D;
            v = (lastu[node] == -1) ? initm[node * MEM_D + dc] : mem[node * MEM_D + dc];
        }
        Ads[row * LDSTR + col] = v;
    }
    __syncthreads();

    const int w    = tid >> 5;        // wave 0..7
    const int lane = tid & 31;
    const int half = lane >> 4;       // 0: lanes 0-15, 1: lanes 16-31
    const int n    = lane & 15;
    const int cb_r = 16 * w;          // r-gate columns
    const int cb_z = MEM_D + 16 * w;  // z-gate columns
    const int cb_n = 2 * MEM_D + 16 * w;

    v8f accR[2]  = {{}, {}};
    v8f accZ[2]  = {{}, {}};
    v8f accN1[2] = {{}, {}};          // i_n part (aggr @ W_ih)
    v8f accN2[2] = {{}, {}};          // h_n part (h @ W_hh)

    // K in [0,512): aggr @ W_ih^T -> r, z, i_n
    for (int k = 0; k < MSG_D; k += 4) {
        int kk = k + 2 * half;
        // A frag (16x4): lane(<16) VGPR0=A[m][k+0], VGPR1=A[m][k+1];
        //                lane(>=16) K offset +2.
        const float2 a0 = *(const float2*)&Ads[n * LDSTR + kk];
        const float2 a1 = *(const float2*)&Ads[(16 + n) * LDSTR + kk];
        v2f A0 = {a0.x, a0.y};
        v2f A1 = {a1.x, a1.y};
        const float2* bprow = (const float2*)&Bp[(kk >> 1) * (2 * GATES)];
        float2 br = bprow[cb_r + n];
        float2 bz = bprow[cb_z + n];
        float2 bn = bprow[cb_n + n];
        v2f B = {br.x, br.y};
        accR[0]  = WMMA_F32(A0, B, accR[0]);
        accR[1]  = WMMA_F32(A1, B, accR[1]);
        B = (v2f){bz.x, bz.y};
        accZ[0]  = WMMA_F32(A0, B, accZ[0]);
        accZ[1]  = WMMA_F32(A1, B, accZ[1]);
        B = (v2f){bn.x, bn.y};
        accN1[0] = WMMA_F32(A0, B, accN1[0]);
        accN1[1] = WMMA_F32(A1, B, accN1[1]);
    }
    // K in [512,640): h @ W_hh^T -> r, z, h_n (h_n kept separate for GRU)
    for (int k = MSG_D; k < KTOT; k += 4) {
        int kk = k + 2 * half;
        const float2 a0 = *(const float2*)&Ads[n * LDSTR + kk];
        const float2 a1 = *(const float2*)&Ads[(16 + n) * LDSTR + kk];
        v2f A0 = {a0.x, a0.y};
        v2f A1 = {a1.x, a1.y};
        const float2* bprow = (const float2*)&Bp[(kk >> 1) * (2 * GATES)];
        float2 br = bprow[cb_r + n];
        float2 bz = bprow[cb_z + n];
        float2 bn = bprow[cb_n + n];
        v2f B = {br.x, br.y};
        accR[0]  = WMMA_F32(A0, B, accR[0]);
        accR[1]  = WMMA_F32(A1, B, accR[1]);
        B = (v2f){bz.x, bz.y};
        accZ[0]  = WMMA_F32(A0, B, accZ[0]);
        accZ[1]  = WMMA_F32(A1, B, accZ[1]);
        B = (v2f){bn.x, bn.y};
        accN2[0] = WMMA_F32(A0, B, accN2[0]);
        accN2[1] = WMMA_F32(A1, B, accN2[1]);
    }

    // GRU epilogue. C/D layout: VGPR v -> row m = 16*tile + v + 8*half, col = 16w + n.
    const int dim = 16 * w + n;
    const float bihr = bih[dim],             bhhr = bhh[dim];
    const float bihz = bih[MEM_D + dim],     bhhz = bhh[MEM_D + dim];
    const float bihn = bih[2 * MEM_D + dim], bhhn = bhh[2 * MEM_D + dim];

    float sq = 0.0f;
#pragma unroll
    for (int tt = 0; tt < 2; ++tt) {
#pragma unroll
        for (int v = 0; v < 8; ++v) {
            int m = 16 * tt + v + 8 * half;
            float h  = Ads[m * LDSTR + MSG_D + dim];
            float r  = sigmoidf(accR[tt][v] + bihr + bhhr);
            float z  = sigmoidf(accZ[tt][v] + bihz + bhhz);
            float nn = tanhf(accN1[tt][v] + bihn + r * (accN2[tt][v] + bhhn));
            float newh = (1.0f - z) * nn + z * h;
            float hn = hasS[m];
            float outv = (hn > 0.0f) ? newh : h;
            float diff = (newh - h) * hn;
            sq += diff * diff;
            outMem[(nodeBase + m) * MEM_D + dim] = outv;
        }
    }
    // wave32 reduction of squared diffs, one atomic per wave
#pragma unroll
    for (int off = 16; off > 0; off >>= 1) sq += __shfl_down(sq, off, 32);
    if (lane == 0) unsafeAtomicAdd(loss, sq);
}

__global__ void k_final(const float* __restrict__ loss, float* __restrict__ out) {
    out[0] = 1.0f * sqrtf(loss[0]) / (float)N_NODES;   // UPDATE_PENALTY = 1.0
}

extern "C" void kernel_launch(void* const* d_in, const int* in_sizes, int n_in,
                              void* d_out, int out_size, void* d_ws, size_t ws_size,
                              hipStream_t stream) {
    const int*   src   = (const int*)d_in[0];
    const int*   dst   = (const int*)d_in[1];
    const int*   prod  = (const int*)d_in[2];
    const int*   t     = (const int*)d_in[3];
    const float* raw   = (const float*)d_in[4];
    const float* mem   = (const float*)d_in[5];
    const int*   lastu = (const int*)d_in[6];
    const float* initm = (const float*)d_in[7];
    const float* Wt    = (const float*)d_in[8];   // (64,1) flat
    const float* bt    = (const float*)d_in[9];
    const float* Wih   = (const float*)d_in[10];  // (384,512)
    const float* Whh   = (const float*)d_in[11];  // (384,128)
    const float* bih   = (const float*)d_in[12];
    const float* bhh   = (const float*)d_in[13];

    // workspace layout
    char* ws = (char*)d_ws;
    size_t off = 0;
    float* sumMsg = (float*)(ws + off); off += (size_t)N_NODES * MSG_D * 4;   // 204.8 MB
    int*   cnt    = (int*)  (ws + off); off += (size_t)N_NODES * 4;
    int*   tmax   = (int*)  (ws + off); off += (size_t)N_NODES * 4;
    float* Bp     = (float*)(ws + off); off += (size_t)KTOT * GATES * 4;
    float* loss   = (float*)(ws + off); off += 16;

    float* outMem  = (float*)d_out;                       // (N, 128)
    float* outLast = outMem + (size_t)N_NODES * MEM_D;    // (N,) as float
    float* outLoss = outLast + N_NODES;                   // scalar

    k_init  <<<2048, 256, 0, stream>>>(sumMsg, cnt, tmax, loss);
    k_wprep <<<(KTOT * GATES + 255) / 256, 256, 0, stream>>>(Wih, Whh, Bp);
    k_events<<<N_EVENTS, 512, 0, stream>>>(src, dst, prod, t, raw, mem, lastu, Wt, bt,
                                           sumMsg, cnt, tmax);
    k_gru   <<<N_NODES / ROWS, 256, 0, stream>>>(sumMsg, cnt, tmax, mem, lastu, initm,
                                                 Bp, bih, bhh, outMem, outLast, loss);
    k_final <<<1, 1, 0, stream>>>(loss, outLoss);
}